// Router_9844065042868
// MI455X (gfx1250) — compile-verified
//
#include <hip/hip_runtime.h>

typedef __attribute__((ext_vector_type(2))) float v2f;
typedef __attribute__((ext_vector_type(4))) float v4f;
typedef __attribute__((ext_vector_type(8))) float v8f;

// Problem geometry (fixed by the reference)
constexpr int DDIM    = 4096;  // D
constexpr int KDIM    = 256;   // KERNEL = D / PATCH
constexpr int NPATCH  = 16;    // PATCH
constexpr int OUTC    = 64;    // OUT
constexpr int ROWS    = 32;    // rows of B per block
constexpr int PSTRIDE = 260;   // LDS row stride (floats); 260 % 64 == 4 -> conflict-free frag reads

__global__ __launch_bounds__(256) void router_fused_kernel(
    const float* __restrict__ x,     // (B, 1, D) f32
    const float* __restrict__ w,     // (OUT, 1, KERNEL) f32
    const float* __restrict__ tie,   // (B, OUT) f32
    float* __restrict__ out)         // (B, OUT) f32
{
    __shared__ float lds_ps[ROWS * PSTRIDE];   // patch_sum tile, 33,280 B

    const int t  = threadIdx.x;
    const int bb = blockIdx.x * ROWS;          // first B-row of this block

    // ---------------- Phase 1: streaming patch-sum into LDS ----------------
    // thread t owns k-quad q (k = 4q..4q+3) and rows rg, rg+4, ..., rg+28
    const int q  = t & 63;
    const int rg = t >> 6;
    #pragma unroll
    for (int r = rg; r < ROWS; r += 4) {
        const float* xp = x + (size_t)(bb + r) * DDIM + 4 * q;
        v4f acc = {0.f, 0.f, 0.f, 0.f};
        #pragma unroll
        for (int p = 0; p < NPATCH; ++p) {
            v4f v = __builtin_nontemporal_load((const v4f*)(xp + p * KDIM));
            acc += v;
        }
        *(v4f*)&lds_ps[r * PSTRIDE + 4 * q] = acc;
    }
    __syncthreads();

    // ---------------- Phase 2: WMMA GEMM  D[m][n] = sum_k ps[m][k] * w[n][k] ----------------
    const int lane  = t & 31;
    const int wave  = t >> 5;        // 8 waves
    const int mtile = wave & 1;      // 2 M-tiles of 16 rows
    const int ntile = wave >> 1;     // 4 N-tiles of 16 cols

    const int l15   = lane & 15;
    const int khalf = (lane >> 4) << 1;            // 0 (lanes 0-15) or 2 (lanes 16-31)
    const int mrow  = mtile * 16 + l15;            // A-fragment row (within LDS tile)
    const int ncol  = ntile * 16 + l15;            // B-fragment column (output channel)

    const float* wrow = w + (size_t)ncol * KDIM;   // w[n][*]

    // Two independent accumulator chains to pipeline the matrix unit
    v8f acc0 = {}; v8f acc1 = {};
    #pragma unroll 4
    for (int k0 = 0; k0 < KDIM; k0 += 8) {
        // A 16x4 f32 layout: lanes 0-15 hold {K0,K1}, lanes 16-31 hold {K2,K3}, M = lane&15
        v2f a0 = *(const v2f*)&lds_ps[mrow * PSTRIDE + k0 + khalf];
        v2f b0 = *(const v2f*)&wrow[k0 + khalf];
        acc0 = __builtin_amdgcn_wmma_f32_16x16x4_f32(
            false, a0, false, b0, (short)0, acc0, false, false);

        v2f a1 = *(const v2f*)&lds_ps[mrow * PSTRIDE + k0 + 4 + khalf];
        v2f b1 = *(const v2f*)&wrow[k0 + 4 + khalf];
        acc1 = __builtin_amdgcn_wmma_f32_16x16x4_f32(
            false, a1, false, b1, (short)0, acc1, false, false);
    }
    v8f acc = acc0 + acc1;

    // ---------------- Epilogue: + tie_noise, store ----------------
    // C/D 16x16 f32 layout: VGPR g -> M = g + 8*(lane>=16), N = lane&15
    const int mhi = (lane >> 4) * 8;
    #pragma unroll
    for (int g = 0; g < 8; ++g) {
        const int row = bb + mtile * 16 + mhi + g;
        const size_t idx = (size_t)row * OUTC + ncol;
        out[idx] = acc[g] + tie[idx];
    }
}

extern "C" void kernel_launch(void* const* d_in, const int* in_sizes, int n_in,
                              void* d_out, int out_size, void* d_ws, size_t ws_size,
                              hipStream_t stream) {
    const float* x   = (const float*)d_in[0];
    const float* w   = (const float*)d_in[1];
    const float* tie = (const float*)d_in[2];
    float* out       = (float*)d_out;

    const int Btot = in_sizes[0] / DDIM;   // 32768
    const int grid = Btot / ROWS;          // 1024 blocks
    router_fused_kernel<<<grid, 256, 0, stream>>>(x, w, tie, out);
}